// Instructor_47639777247870
// MI455X (gfx1250) — compile-verified
//
#include <hip/hip_runtime.h>
#include <math.h>

#define NN 256
#define BB 64

typedef __attribute__((ext_vector_type(2))) float v2f;
typedef __attribute__((ext_vector_type(8))) float v8f;

__device__ __forceinline__ int refl(int k) {
    k = (k < 0) ? -k : k;
    k = (k > NN - 1) ? (2 * (NN - 1) - k) : k;
    return k;
}

#define DEF_OFF_C (1.0f + 2.0f / 255.0f)
#define PW_C      (2.0f / 256.0f)

// ---------------------------------------------------------------------------
// Kernel 1: separable 5x5 Gaussian conv (reflect pad) + directional diff + relu
// tile 32x32 per block, one (batch, channel) per block.
// accu_x[i,j] = relu( s(i,j) - s(i,j-1) )   (s includes iden + DEF_OFF)
// accu_y[i,j] = relu( s(i,j) - s(i-1,j) )
// ---------------------------------------------------------------------------
__global__ __launch_bounds__(256) void k_conv_diff(
    const float* __restrict__ pg, float* __restrict__ accuX,
    float* __restrict__ accuY, float w0, float w1, float w2)
{
    __shared__ float patch[37 * 38];   // rows i0-3..i0+33, cols j0-3..j0+33
    __shared__ float hbuf [37 * 34];   // horizontal pass, cols j_local -1..31
    __shared__ float sbuf [33 * 34];   // vertical pass, rows i_local -1..31

    const float w[5] = {w0, w1, w2, w1, w0};
    int tile = blockIdx.x & 63;
    int ch   = (blockIdx.x >> 6) & 1;
    int b    = blockIdx.x >> 7;
    int i0   = (tile >> 3) * 32;
    int j0   = (tile & 7) * 32;
    const float* src = pg + (size_t)(b * 2 + ch) * (NN * NN);
    int tid = threadIdx.x;

    for (int idx = tid; idx < 37 * 37; idx += 256) {
        int pr = idx / 37, pc = idx % 37;
        patch[pr * 38 + pc] = src[refl(i0 + pr - 3) * NN + refl(j0 + pc - 3)];
    }
    __syncthreads();

    for (int idx = tid; idx < 37 * 33; idx += 256) {
        int r = idx / 33, c = idx % 33;
        const float* p = &patch[r * 38 + c];
        hbuf[r * 34 + c] = w[0]*p[0] + w[1]*p[1] + w[2]*p[2] + w[3]*p[3] + w[4]*p[4];
    }
    __syncthreads();

    for (int idx = tid; idx < 33 * 33; idx += 256) {
        int r = idx / 33, c = idx % 33;
        sbuf[r * 34 + c] = w[0]*hbuf[(r+0)*34 + c] + w[1]*hbuf[(r+1)*34 + c]
                         + w[2]*hbuf[(r+2)*34 + c] + w[3]*hbuf[(r+3)*34 + c]
                         + w[4]*hbuf[(r+4)*34 + c];
    }
    __syncthreads();

    float* dst = (ch ? accuY : accuX) + (size_t)b * (NN * NN);
    #pragma unroll
    for (int q = 0; q < 4; ++q) {
        int p  = tid + 256 * q;
        int li = p >> 5, lj = p & 31;
        int i = i0 + li, j = j0 + lj;
        float sc = sbuf[(li + 1) * 34 + (lj + 1)];
        float val;
        if (ch == 0) {
            if (j == 0) val = sc - 1.0f + DEF_OFF_C;                 // seq(0) = -1
            else        val = (sc - sbuf[(li + 1) * 34 + lj]) + 2.0f / 255.0f;
        } else {
            if (i == 0) val = sc - 1.0f + DEF_OFF_C;
            else        val = (sc - sbuf[li * 34 + (lj + 1)]) + 2.0f / 255.0f;
        }
        dst[i * NN + j] = fmaxf(val, 0.0f);
    }
}

// ---------------------------------------------------------------------------
// WMMA prefix-scan core: inclusive cumsum of a 16x256 strip held in LDS
// (stride 257), using V_WMMA_F32_16X16X4_F32 against triangular ones
// matrices, plus per-lane inter-tile carry broadcast from lane 15/31.
// ---------------------------------------------------------------------------
// A(16x4): lanes 0-15 hold {K0,K1}, lanes 16-31 hold {K2,K3}, M = lane&15
// B(4x16): lanes 0-15 hold {K0,K1}, lanes 16-31 hold {K2,K3}, N = lane&15
// C/D(16x16): VGPR v -> M = v (lanes 0-15) / v+8 (lanes 16-31), N = lane&15

// ---------------------------------------------------------------------------
// Kernel 2x: cumsum along W + clip -> gx (in place over accuX: block stages
// its entire input in LDS before any global write).
// ---------------------------------------------------------------------------
__global__ __launch_bounds__(32) void k_scan_x(const float* __restrict__ accu,
                                               float* __restrict__ gx)
{
    __shared__ float lds[16 * 257];
    int b    = blockIdx.x >> 4;
    int r0   = (blockIdx.x & 15) * 16;
    int lane = threadIdx.x;

    const float* src = accu + (size_t)b * 65536 + (size_t)r0 * 256;
    for (int r = 0; r < 16; ++r) {
        const float4* s4 = (const float4*)(src + r * 256) + lane * 2;
        float4 u = s4[0], v4 = s4[1];
        float* d = &lds[r * 257 + lane * 8];
        d[0]=u.x; d[1]=u.y; d[2]=u.z; d[3]=u.w;
        d[4]=v4.x; d[5]=v4.y; d[6]=v4.z; d[7]=v4.w;
    }
    __syncthreads();

    int hi  = (lane >= 16);
    int m   = lane & 15;
    int n   = lane & 15;
    int klo = hi ? 2 : 0;

    v2f bf[4];
    #pragma unroll
    for (int kb = 0; kb < 4; ++kb) {
        int kk = kb * 4 + klo;
        bf[kb].x = (kk     <= n) ? 1.0f : 0.0f;
        bf[kb].y = (kk + 1 <= n) ? 1.0f : 0.0f;
    }

    float carry[8] = {0, 0, 0, 0, 0, 0, 0, 0};
    float* dst = gx + (size_t)b * 65536;

    for (int t = 0; t < 16; ++t) {
        v8f p = {};
        #pragma unroll
        for (int kb = 0; kb < 4; ++kb) {
            v2f a;
            int k0 = t * 16 + kb * 4 + klo;
            a.x = lds[m * 257 + k0];
            a.y = lds[m * 257 + k0 + 1];
            p = __builtin_amdgcn_wmma_f32_16x16x4_f32(false, a, false, bf[kb],
                                                      (short)0, p, false, false);
        }
        int   col  = t * 16 + n;
        float iden = (2.0f * col - 255.0f) * (1.0f / 255.0f);
        #pragma unroll
        for (int v = 0; v < 8; ++v) {
            float pv  = p[v];
            float tot = __shfl(pv, 15, 16);          // tile total per row
            float val = pv + carry[v];
            carry[v] += tot;
            float samp = val - DEF_OFF_C;
            float prim = fminf(fmaxf(samp - iden, -PW_C), PW_C);
            float g    = fminf(fmaxf(prim + iden, -1.0f), 1.0f);
            int row = r0 + v + (hi ? 8 : 0);
            dst[row * 256 + col] = g;
        }
    }
}

// ---------------------------------------------------------------------------
// Kernel 2y: cumsum along H + clip -> gy (in place over accuY). Each wave
// owns 16 columns, transposed into LDS; WMMA scan identical to k_scan_x;
// results written back in place into LDS, then coalesced 128B row stores.
// ---------------------------------------------------------------------------
__global__ __launch_bounds__(64) void k_scan_y(const float* __restrict__ accu,
                                               float* __restrict__ gy)
{
    __shared__ float lds[2 * 16 * 257];
    int b    = blockIdx.x >> 3;
    int cblk = (blockIdx.x & 7) * 32;
    int tid  = threadIdx.x;
    int w    = tid >> 5;
    int lane = tid & 31;
    int c0   = cblk + w * 16;
    float* my = &lds[w * 16 * 257];
    const float* src = accu + (size_t)b * 65536;

    for (int r = 0; r < 256; r += 2) {
        int row = r + (lane >> 4);
        int col = c0 + (lane & 15);
        my[(lane & 15) * 257 + row] = src[row * 256 + col];
    }
    __syncthreads();

    int hi  = (lane >= 16);
    int m   = lane & 15;       // local column (transposed row)
    int n   = lane & 15;       // image row within tile
    int klo = hi ? 2 : 0;

    v2f bf[4];
    #pragma unroll
    for (int kb = 0; kb < 4; ++kb) {
        int kk = kb * 4 + klo;
        bf[kb].x = (kk     <= n) ? 1.0f : 0.0f;
        bf[kb].y = (kk + 1 <= n) ? 1.0f : 0.0f;
    }

    float carry[8] = {0, 0, 0, 0, 0, 0, 0, 0};

    for (int t = 0; t < 16; ++t) {
        v8f p = {};
        #pragma unroll
        for (int kb = 0; kb < 4; ++kb) {
            v2f a;
            int k0 = t * 16 + kb * 4 + klo;
            a.x = my[m * 257 + k0];
            a.y = my[m * 257 + k0 + 1];
            p = __builtin_amdgcn_wmma_f32_16x16x4_f32(false, a, false, bf[kb],
                                                      (short)0, p, false, false);
        }
        int   irow = t * 16 + n;                      // image row
        float iden = (2.0f * irow - 255.0f) * (1.0f / 255.0f);
        #pragma unroll
        for (int v = 0; v < 8; ++v) {
            float pv  = p[v];
            float tot = __shfl(pv, 15, 16);
            float val = pv + carry[v];
            carry[v] += tot;
            float samp = val - DEF_OFF_C;
            float prim = fminf(fmaxf(samp - iden, -PW_C), PW_C);
            float g    = fminf(fmaxf(prim + iden, -1.0f), 1.0f);
            // in-place: columns t*16..t*16+15 already consumed for this tile,
            // future tiles read strictly larger column indices.
            my[(v + (hi ? 8 : 0)) * 257 + irow] = g;
        }
    }
    __syncthreads();

    // coalesced write-out: 64 threads = 2 rows x 32 contiguous cols (128B/row)
    float* dst = gy + (size_t)b * 65536 + cblk;
    int c = tid & 31;
    const float* rd = &lds[(c >> 4) * (16 * 257) + (c & 15) * 257];
    for (int i = tid >> 5; i < 256; i += 2)
        dst[i * 256 + c] = rd[i];
}

// ---------------------------------------------------------------------------
// Kernel 3: bilinear grid-sample. Coordinates are within 1 pixel of identity,
// so reads are local and cache-resident (entire problem fits in 192MB L2).
// ---------------------------------------------------------------------------
__global__ __launch_bounds__(256) void k_sample(
    const float* __restrict__ img, const float* __restrict__ gx,
    const float* __restrict__ gy, float* __restrict__ out)
{
    int b = blockIdx.x >> 8;
    int i = blockIdx.x & 255;
    int j = threadIdx.x;
    size_t gidx = (size_t)b * 65536 + (size_t)i * 256 + j;

    float fx = gx[gidx], fy = gy[gidx];
    float ix = (fx + 1.0f) * 0.5f * 255.0f;   // in [0,255]
    float iy = (fy + 1.0f) * 0.5f * 255.0f;
    float x0 = floorf(ix), y0 = floorf(iy);
    float wx = ix - x0,    wy = iy - y0;

    int X0 = (int)x0; if (X0 < 0) X0 = 0; if (X0 > 255) X0 = 255;
    int Y0 = (int)y0; if (Y0 < 0) Y0 = 0; if (Y0 > 255) Y0 = 255;
    float v10 = (x0 + 1.0f <= 255.0f) ? 1.0f : 0.0f;
    float v01 = (y0 + 1.0f <= 255.0f) ? 1.0f : 0.0f;
    int X1 = X0 + 1; if (X1 > 255) X1 = 255;
    int Y1 = Y0 + 1; if (Y1 > 255) Y1 = 255;

    float w00 = (1.0f - wx) * (1.0f - wy);
    float w10 = wx * (1.0f - wy) * v10;
    float w01 = (1.0f - wx) * wy * v01;
    float w11 = wx * wy * v10 * v01;

    const float* base = img + (size_t)b * 3 * 65536;
    float* ob = out + (size_t)b * 3 * 65536 + (size_t)i * 256 + j;
    #pragma unroll
    for (int c = 0; c < 3; ++c) {
        const float* pc = base + c * 65536;
        float r00 = pc[Y0 * 256 + X0];
        float r10 = pc[Y0 * 256 + X1];
        float r01 = pc[Y1 * 256 + X0];
        float r11 = pc[Y1 * 256 + X1];
        ob[c * 65536] = r00 * w00 + r10 * w10 + r01 * w01 + r11 * w11;
    }
}

// ---------------------------------------------------------------------------
extern "C" void kernel_launch(void* const* d_in, const int* in_sizes, int n_in,
                              void* d_out, int out_size, void* d_ws, size_t ws_size,
                              hipStream_t stream)
{
    const float* image = (const float*)d_in[0];   // (64,3,256,256) f32
    const float* pg    = (const float*)d_in[1];   // (64,2,256,256) f32
    float* out = (float*)d_out;                   // (64,3,256,256) f32

    // workspace: accuX | accuY ; scans run in place -> gx = accuX, gy = accuY
    float* accuX = (float*)d_ws;                          // 16.78 MB
    float* accuY = accuX + (size_t)BB * NN * NN;          // 16.78 MB
    float* gx = accuX;
    float* gy = accuY;

    // Gaussian taps: sigma = 5*0.15 + 0.35 = 1.1
    float sigma = 5.0f * 0.15f + 0.35f;
    float p0 = expf(-0.5f * (2.0f / sigma) * (2.0f / sigma));
    float p1 = expf(-0.5f * (1.0f / sigma) * (1.0f / sigma));
    float p2 = 1.0f;
    float s  = 2.0f * p0 + 2.0f * p1 + p2;
    float w0 = p0 / s, w1 = p1 / s, w2 = p2 / s;

    k_conv_diff<<<dim3(BB * 2 * 64), dim3(256), 0, stream>>>(pg, accuX, accuY, w0, w1, w2);
    k_scan_x  <<<dim3(BB * 16),      dim3(32),  0, stream>>>(accuX, gx);
    k_scan_y  <<<dim3(BB * 8),       dim3(64),  0, stream>>>(accuY, gy);
    k_sample  <<<dim3(BB * 256),     dim3(256), 0, stream>>>(image, gx, gy, out);
}